// ROCKET_75067438400169
// MI455X (gfx1250) — compile-verified
//
#include <hip/hip_runtime.h>
#include <hip/hip_bf16.h>

// ROCKET-style multi-kernel conv + [PPV, MPV, MAXPV] pooling for MI455X (gfx1250).
// Implicit-GEMM conv via V_WMMA_F32_16X16X4_F32 (wave32) over a zero-padded input.
// Biased (non-negative) im2col deltas + masked unsigned index -> SADDR-form
// global loads with zero per-load VALU; each B fragment feeds 2 M-tiles.

typedef float v2f __attribute__((ext_vector_type(2)));
typedef float v8f __attribute__((ext_vector_type(8)));
typedef int   v2i __attribute__((ext_vector_type(2)));

#define IMG   224
#define NPIX  (IMG * IMG)          // 50176
#define PAD   4                    // halo covers k=9 (p=4)
#define PW    (IMG + 2 * PAD)      // 232
#define PH    (IMG + 2 * PAD)      // 232
#define PP    (PW * PH)            // 53824 (per channel, padded)
#define BIAS_OFF (PAD * PW + PAD)  // 932: makes all im2col deltas non-negative
#define IDX_MASK 0x1FFFF           // identity mask (max index ~109.5k < 131072)
#define COUT  96
#define CIN   3
#define NTILES (NPIX / 16)         // 3136 pixel tiles of 16 (each within one row)
#define WAVES_PER_BLOCK 8
#define TILES_PER_WAVE  4
#define TILES_PER_BLOCK (WAVES_PER_BLOCK * TILES_PER_WAVE)  // 32
#define CHUNKS (NTILES / TILES_PER_BLOCK)                   // 98 (exact)
#define NB    8
#define MGROUPS 3                  // 3 groups of 32 channels
#define NGROUPS 4
#define ACC_FLOATS (NGROUPS * NB * COUT * 3)  // 9216
#define PAD_TOTAL (NB * CIN * PP)             // 1,291,776 floats

static_assert(NTILES == TILES_PER_BLOCK * CHUNKS, "tile decomposition must be exact");
static_assert(IMG % 16 == 0, "tiles must not cross image rows");

__global__ void ROCKET_zero_acc(float* acc, int n) {
    int i = blockIdx.x * blockDim.x + threadIdx.x;
    if (i < n) acc[i] = 0.0f;   // max slot: relu output >= 0, so 0.0f init is correct
}

// X [8,3,224,224] -> Xp [8,3,232,232] with 4-px zero halo (rebuilt every call).
__global__ void ROCKET_pad(const float* __restrict__ X, float* __restrict__ Xp, int total) {
    int i = blockIdx.x * blockDim.x + threadIdx.x;
    if (i >= total) return;
    int col = i % PW;  int t = i / PW;
    int row = t % PH;  t /= PH;          // t = b*CIN + ci
    int x = col - PAD, y = row - PAD;
    float v = 0.0f;
    if (((unsigned)x < IMG) && ((unsigned)y < IMG))
        v = X[(size_t)t * NPIX + y * IMG + x];
    Xp[i] = v;
}

template <int K>
__global__ __launch_bounds__(256) void ROCKET_conv_pool(
    const float* __restrict__ Xp,    // padded input [8,3,232,232]
    const float* __restrict__ W,     // [96,3,1,K,K]
    const float* __restrict__ Bias,  // [96]
    float* __restrict__ acc,         // [4,8,96,3] (cnt,sum,max)
    int g)
{
    constexpr int KK  = CIN * K * K;
    constexpr int KK4 = (KK + 3) & ~3;      // pad K-dim to multiple of 4 (zeros in A)
    constexpr int P   = (K - 1) / 2;

    __shared__ float sW[32 * KK4];          // 32-channel weight tile, M-major, zero padded
    __shared__ int   sOff[KK4];             // BIASED im2col delta (always >= 0)
    __shared__ float sB[32];

    int bid   = blockIdx.x;
    int chunk = bid % CHUNKS;   bid /= CHUNKS;
    int mt    = bid % MGROUPS;  int b = bid / MGROUPS;
    int mbase = mt * 32;

    for (int i = threadIdx.x; i < 32 * KK4; i += blockDim.x) {
        int m = i / KK4, kk = i % KK4;
        sW[i] = (kk < KK) ? W[(mbase + m) * KK + kk] : 0.0f;
    }
    if (threadIdx.x < 32) sB[threadIdx.x] = Bias[mbase + threadIdx.x];
    for (int kk = threadIdx.x; kk < KK4; kk += blockDim.x) {
        int kc  = (kk < KK) ? kk : 0;       // padded slots: A==0, value irrelevant
        int ci  = kc / (K * K);
        int rem = kc % (K * K);
        int dy  = rem / K - P, dx = rem % K - P;
        sOff[kk] = ci * PP + dy * PW + dx + BIAS_OFF;   // >= 0 by construction
    }
    __syncthreads();

    const int lane = threadIdx.x & 31;
    const int wv   = threadIdx.x >> 5;
    const int half = lane >> 4;             // A/B: K half; C/D: M half
    const int l15  = lane & 15;

    const float* Xb = Xp + (size_t)b * CIN * PP;
    int tile0 = chunk * TILES_PER_BLOCK + wv * TILES_PER_WAVE;

    // Wave-uniform per-tile base pointers (bias pre-subtracted, still >= Xp start).
    const float* tb[TILES_PER_WAVE];
#pragma unroll
    for (int t = 0; t < TILES_PER_WAVE; ++t) {
        int pix0 = (tile0 + t) * 16;        // tile start pixel (row-aligned chunk of 16)
        int py = pix0 / IMG, pxb = pix0 % IMG;
        int sidx = __builtin_amdgcn_readfirstlane(
                       (py + PAD) * PW + pxb + PAD - BIAS_OFF);  // >= 0 (min is 0)
        tb[t] = Xb + sidx;
        __builtin_prefetch(tb[t] + BIAS_OFF + l15, 0, 3);   // global_prefetch_b8
    }

    v8f c[2][TILES_PER_WAVE];
#pragma unroll
    for (int j = 0; j < 2; ++j)
#pragma unroll
        for (int t = 0; t < TILES_PER_WAVE; ++t) c[j][t] = (v8f){};

    const float* sW0 = sW + l15 * KK4;          // A row for M-tile 0 (M = l15)
    const float* sW1 = sW + (l15 + 16) * KK4;   // A row for M-tile 1 (M = l15 + 16)

#pragma unroll 2
    for (int ks = 0; ks < KK4 / 4; ++ks) {
        int k0 = ks * 4 + 2 * half;         // 8-byte aligned (k0 even)
        v2f a0 = *(const v2f*)(sW0 + k0);   // ds_load_b64: A frag, channels [0,16)
        v2f a1 = *(const v2f*)(sW1 + k0);   // ds_load_b64: A frag, channels [16,32)
        v2i o  = *(const v2i*)(sOff + k0);  // ds_load_b64: biased im2col deltas
        // Masked => provably-unsigned 32-bit index => SADDR + voffset + scale_offset.
        unsigned i0 = (unsigned)(o.x + l15) & IDX_MASK;   // mask is identity
        unsigned i1 = (unsigned)(o.y + l15) & IDX_MASK;
#pragma unroll
        for (int t = 0; t < TILES_PER_WAVE; ++t) {
            v2f bb;
            bb.x = tb[t][i0];               // global_load_b32 v, v_idx, s[..] scale_offset
            bb.y = tb[t][i1];
            c[0][t] = __builtin_amdgcn_wmma_f32_16x16x4_f32(
                          false, a0, false, bb, (short)0, c[0][t], false, false);
            c[1][t] = __builtin_amdgcn_wmma_f32_16x16x4_f32(
                          false, a1, false, bb, (short)0, c[1][t], false, false);
        }
    }

    // bias + ReLU + fused pooling stats; element (m,n): m = 16*j + r + 8*half, n = l15
#pragma unroll
    for (int j = 0; j < 2; ++j) {
        float bias[8];
#pragma unroll
        for (int r = 0; r < 8; ++r) bias[r] = sB[16 * j + r + 8 * half];

        float cnt[8], sum[8], mx[8];
#pragma unroll
        for (int r = 0; r < 8; ++r) { cnt[r] = 0.0f; sum[r] = 0.0f; mx[r] = 0.0f; }
#pragma unroll
        for (int t = 0; t < TILES_PER_WAVE; ++t) {
#pragma unroll
            for (int r = 0; r < 8; ++r) {
                float v = c[j][t][r] + bias[r];
                v = fmaxf(v, 0.0f);
                cnt[r] += (v > 0.0f) ? 1.0f : 0.0f;
                sum[r] += v;
                mx[r]   = fmaxf(mx[r], v);
            }
        }

        // Butterfly-reduce across the 16 lanes sharing each channel.
#pragma unroll
        for (int off = 1; off < 16; off <<= 1) {
#pragma unroll
            for (int r = 0; r < 8; ++r) {
                cnt[r] += __shfl_xor(cnt[r], off, 32);
                sum[r] += __shfl_xor(sum[r], off, 32);
                mx[r]   = fmaxf(mx[r], __shfl_xor(mx[r], off, 32));
            }
        }

        if (l15 == 0) {
#pragma unroll
            for (int r = 0; r < 8; ++r) {
                int ch = mbase + 16 * j + r + 8 * half;
                float* p = acc + (((size_t)g * NB + b) * COUT + ch) * 3;
                atomicAdd(p + 0, cnt[r]);
                atomicAdd(p + 1, sum[r]);
                atomicMax((unsigned int*)(p + 2), __float_as_uint(mx[r]));
            }
        }
    }
}

__global__ void ROCKET_finalize(const float* __restrict__ acc, float* __restrict__ out) {
    int i = blockIdx.x * blockDim.x + threadIdx.x;
    if (i >= NB * NGROUPS * 3 * COUT) return;
    int c = i % COUT;  int t = i / COUT;
    int kind = t % 3;  t /= 3;
    int g = t % NGROUPS;  int b = t / NGROUPS;
    const float* p = acc + (((size_t)g * NB + b) * COUT + c) * 3;
    float cnt = p[0], sum = p[1], mx = p[2];
    float v;
    if (kind == 0)      v = cnt / (float)NPIX;
    else if (kind == 1) v = (cnt > 0.0f) ? (sum / cnt) : 0.0f;
    else                v = mx;
    out[(size_t)b * (NGROUPS * 3 * COUT) + g * (3 * COUT) + kind * COUT + c] = v;
}

extern "C" void kernel_launch(void* const* d_in, const int* in_sizes, int n_in,
                              void* d_out, int out_size, void* d_ws, size_t ws_size,
                              hipStream_t stream) {
    const float* X  = (const float*)d_in[0];
    const float* w3 = (const float*)d_in[1]; const float* b3 = (const float*)d_in[2];
    const float* w5 = (const float*)d_in[3]; const float* b5 = (const float*)d_in[4];
    const float* w7 = (const float*)d_in[5]; const float* b7 = (const float*)d_in[6];
    const float* w9 = (const float*)d_in[7]; const float* b9 = (const float*)d_in[8];

    float* acc = (float*)d_ws;                           // 9216 floats (36 KB)
    float* Xp  = (float*)((char*)d_ws + 65536);          // padded input, ~5.2 MB
    float* out = (float*)d_out;

    ROCKET_zero_acc<<<(ACC_FLOATS + 255) / 256, 256, 0, stream>>>(acc, ACC_FLOATS);
    ROCKET_pad<<<(PAD_TOTAL + 255) / 256, 256, 0, stream>>>(X, Xp, PAD_TOTAL);

    int nb = NB * MGROUPS * CHUNKS;          // 2352 blocks per group
    ROCKET_conv_pool<3><<<nb, 256, 0, stream>>>(Xp, w3, b3, acc, 0);
    ROCKET_conv_pool<5><<<nb, 256, 0, stream>>>(Xp, w5, b5, acc, 1);
    ROCKET_conv_pool<7><<<nb, 256, 0, stream>>>(Xp, w7, b7, acc, 2);
    ROCKET_conv_pool<9><<<nb, 256, 0, stream>>>(Xp, w9, b9, acc, 3);

    ROCKET_finalize<<<(NB * NGROUPS * 3 * COUT + 255) / 256, 256, 0, stream>>>(acc, out);
}